// GCC_79293686219267
// MI455X (gfx1250) — compile-verified
//
#include <hip/hip_runtime.h>

typedef unsigned int u32;
typedef __attribute__((ext_vector_type(2))) float v2f;
typedef __attribute__((ext_vector_type(8))) float v8f;

#define B_    8192
#define D_    768
#define G_    24576
#define O_    3072
#define TOPK  64
#define AUXK  256
#define EPSF  1e-5f
#define NTHR  256
#define SEG   (G_ / NTHR)   // 96 indices per thread, 96%32==0 -> 3 bitmap words

// ---------- helpers ----------
__device__ __forceinline__ u32 f2key(float f) {
    // order-preserving map: larger float -> larger key
    u32 u = __float_as_uint(f);
    return (u & 0x80000000u) ? ~u : (u | 0x80000000u);
}

// key for selection; dead==nullptr -> raw preact, else masked (preact*dead_mask)
__device__ __forceinline__ u32 load_key(const float* __restrict__ row,
                                        const u32* __restrict__ dead, int i) {
    float f = row[i];
    if (dead) f = dead[i] ? f : 0.0f;
    if (f == 0.0f) f = 0.0f;   // canonicalize -0 -> +0 (ties at zero compare equal)
    return f2key(f);
}

// Block-wide radix select of the k-th largest key among n candidates.
// Returns threshold key and count of keys strictly greater than it.
__device__ void radix_select(const float* __restrict__ row, const u32* __restrict__ dead,
                             u32* hist, u32* bc, int n, int k,
                             u32* thresh_out, int* greater_out) {
    u32 prefix = 0;
    int remaining = k;
    int greater = 0;
    for (int level = 3; level >= 0; --level) {
        const int shift = level * 8;
        for (int i = threadIdx.x; i < 256; i += NTHR) hist[i] = 0;
        __syncthreads();
        const u32 maskHi = (level == 3) ? 0u : (0xFFFFFFFFu << (8 * (level + 1)));
        for (int i = threadIdx.x; i < n; i += NTHR) {
            u32 kv = load_key(row, dead, i);
            if ((kv & maskHi) == prefix)
                atomicAdd(&hist[(kv >> shift) & 255u], 1u);
        }
        __syncthreads();
        if (threadIdx.x == 0) {
            int cum = 0, bin = 255;
            for (; bin > 0; --bin) {
                int c = (int)hist[bin];
                if (cum + c >= remaining) break;
                cum += c;
            }
            bc[0] = (u32)bin;
            bc[1] = (u32)cum;
        }
        __syncthreads();
        u32 bin = bc[0];
        int cum = (int)bc[1];
        prefix |= bin << shift;
        greater += cum;
        remaining -= cum;
        __syncthreads();
    }
    *thresh_out = prefix;
    *greater_out = greater;
}

// serial exclusive scan over NTHR entries (tiny, deterministic)
__device__ __forceinline__ void excl_scan(u32* tsc) {
    __syncthreads();
    if (threadIdx.x == 0) {
        u32 a = 0;
        for (int i = 0; i < NTHR; ++i) { u32 c = tsc[i]; tsc[i] = a; a += c; }
    }
    __syncthreads();
}

// ---------- kernels ----------
__global__ __launch_bounds__(256) void k_init(u32* active, int* counter) {
    int i = blockIdx.x * 256 + threadIdx.x;
    if (i < G_) active[i] = 0u;
    if (i == 0) *counter = 0;
}

__global__ __launch_bounds__(256)
void k_layernorm(const float* __restrict__ x, float* __restrict__ xn,
                 float* __restrict__ mu_o, float* __restrict__ sd_o) {
    __shared__ float red[256];
    const int b = blockIdx.x, t = threadIdx.x;
    const float* row = x + (size_t)b * D_;
    float s = 0.f;
    for (int i = t; i < D_; i += 256) s += row[i];
    red[t] = s; __syncthreads();
    for (int o = 128; o > 0; o >>= 1) { if (t < o) red[t] += red[t + o]; __syncthreads(); }
    const float mu = red[0] / (float)D_;
    __syncthreads();
    float s2 = 0.f;
    for (int i = t; i < D_; i += 256) { float d = row[i] - mu; s2 += d * d; }
    red[t] = s2; __syncthreads();
    for (int o = 128; o > 0; o >>= 1) { if (t < o) red[t] += red[t + o]; __syncthreads(); }
    const float sd  = sqrtf(red[0] / (float)(D_ - 1));   // unbiased (ddof=1), torch-style
    const float inv = 1.0f / (sd + EPSF);
    for (int i = t; i < D_; i += 256) xn[(size_t)b * D_ + i] = (row[i] - mu) * inv;
    if (t == 0) { mu_o[b] = mu; sd_o[b] = sd; }
}

// preact[B,G] = xn[B,D] @ Wenc[D,G] + b_enc, fp32 WMMA (exact precision path)
// block tile 128x128, 8 waves (2M x 4N), each wave 64x32 = 4x2 tiles of 16x16
__global__ __launch_bounds__(256)
void k_encoder(const float* __restrict__ xn, const float* __restrict__ Wenc,
               const float* __restrict__ b_enc, float* __restrict__ preact) {
    const int lane  = threadIdx.x & 31;
    const int wave  = threadIdx.x >> 5;
    const int waveM = wave & 1, waveN = wave >> 1;
    const int hh    = lane >> 4;       // half-wave
    const int lm    = lane & 15;
    const int rb = blockIdx.y * 128 + waveM * 64;
    const int cb = blockIdx.x * 128 + waveN * 32;

    v8f acc[4][2];
#pragma unroll
    for (int i = 0; i < 4; ++i)
#pragma unroll
        for (int j = 0; j < 2; ++j)
            acc[i][j] = (v8f){0.f,0.f,0.f,0.f,0.f,0.f,0.f,0.f};

    for (int kk = 0; kk < D_; kk += 4) {
        const int k0 = kk + hh * 2;   // lanes 0-15 hold K={k0,k0+1}, lanes 16-31 K={k0+2,k0+3}
        v2f a[4], bfr[2];
#pragma unroll
        for (int mt = 0; mt < 4; ++mt)
            a[mt] = *(const v2f*)(xn + (size_t)(rb + mt * 16 + lm) * D_ + k0);
#pragma unroll
        for (int nt = 0; nt < 2; ++nt) {
            const int col = cb + nt * 16 + lm;
            bfr[nt].x = Wenc[(size_t)k0 * G_ + col];
            bfr[nt].y = Wenc[(size_t)(k0 + 1) * G_ + col];
        }
#pragma unroll
        for (int mt = 0; mt < 4; ++mt)
#pragma unroll
            for (int nt = 0; nt < 2; ++nt)
                acc[mt][nt] = __builtin_amdgcn_wmma_f32_16x16x4_f32(
                    false, a[mt], false, bfr[nt], (short)0, acc[mt][nt], false, false);
    }

#pragma unroll
    for (int nt = 0; nt < 2; ++nt) {
        const int col = cb + nt * 16 + lm;
        const float be = b_enc[col];
#pragma unroll
        for (int mt = 0; mt < 4; ++mt)
#pragma unroll
            for (int v = 0; v < 8; ++v) {
                const int m = rb + mt * 16 + hh * 8 + v;
                preact[(size_t)m * G_ + col] = acc[mt][nt][v] + be;
            }
    }
}

// per-row exact top-64: compact (idx,relu(val)) list, active flags, selection bitmap
__global__ __launch_bounds__(256)
void k_top64(const float* __restrict__ preact, u32* __restrict__ active,
             u32* __restrict__ bitmap, u32* __restrict__ fidx, float* __restrict__ fval) {
    __shared__ u32 hist[256];
    __shared__ u32 tsc[256];
    __shared__ u32 bc[4];
    const int b = blockIdx.x, t = threadIdx.x;
    const float* row = preact + (size_t)b * G_;

    u32 thresh; int greater;
    radix_select(row, nullptr, hist, bc, G_, TOPK, &thresh, &greater);
    const int needEq = TOPK - greater;

    const int base = t * SEG;
    int eqCnt = 0;
    for (int i = 0; i < SEG; ++i) eqCnt += (load_key(row, nullptr, base + i) == thresh);
    tsc[t] = (u32)eqCnt; excl_scan(tsc);
    const int eqBase = (int)tsc[t];
    __syncthreads();

    int selCnt = 0;
    { int e = eqBase;
      for (int i = 0; i < SEG; ++i) {
          u32 kv = load_key(row, nullptr, base + i);
          if (kv > thresh) selCnt++;
          else if (kv == thresh) { if (e < needEq) selCnt++; e++; }
      } }
    tsc[t] = (u32)selCnt; excl_scan(tsc);
    int slot = (int)tsc[t];

    u32 w[SEG / 32] = {0u, 0u, 0u};
    { int e = eqBase;
      for (int i = 0; i < SEG; ++i) {
          u32 kv = load_key(row, nullptr, base + i);
          bool sel = false;
          if (kv > thresh) sel = true;
          else if (kv == thresh) { sel = (e < needEq); e++; }
          if (sel) {
              const float f = row[base + i];
              fidx[(size_t)b * TOPK + slot] = (u32)(base + i);
              fval[(size_t)b * TOPK + slot] = fmaxf(f, 0.0f);
              slot++;
              if (f > 0.0f) {                 // feats nonzero here
                  active[base + i] = 1u;      // benign race: all write 1
                  w[i >> 5] |= 1u << (i & 31);
              }
          }
      } }
    u32* bmrow = bitmap + (size_t)b * (G_ / 32);
#pragma unroll
    for (int j = 0; j < SEG / 32; ++j) bmrow[t * (SEG / 32) + j] = w[j];
}

__global__ __launch_bounds__(256)
void k_dead(const u32* __restrict__ active, const int* __restrict__ stats,
            const int* __restrict__ thr, u32* __restrict__ dead) {
    const int g = blockIdx.x * 256 + threadIdx.x;
    if (g >= G_) return;
    const int T = *thr;
    // new_stats = stats*all_zero + 1 ; dead = new_stats > T
    const int ns = active[g] ? 1 : (stats[g] + 1);
    dead[g] = (ns > T) ? 1u : 0u;
}

// per-row top-256 over masked preact; also counts nonzero dead acts and
// rewrites the preact row in-place into the final sparse feats row.
__global__ __launch_bounds__(256)
void k_aux(float* __restrict__ preact, const u32* __restrict__ dead,
           const u32* __restrict__ bitmap, u32* __restrict__ aidx,
           float* __restrict__ aval, int* __restrict__ counter) {
    __shared__ u32 hist[256];
    __shared__ u32 tsc[256];
    __shared__ u32 bc[4];
    const int b = blockIdx.x, t = threadIdx.x;
    float* row = preact + (size_t)b * G_;

    // num_dead contribution: count(dead && preact != 0) for this row
    int nz = 0;
    for (int i = t; i < G_; i += 256) {
        if (dead[i] && row[i] != 0.0f) nz++;
    }
    tsc[t] = (u32)nz; __syncthreads();
    for (int o = 128; o > 0; o >>= 1) { if (t < o) tsc[t] += tsc[t + o]; __syncthreads(); }
    if (t == 0) atomicAdd(counter, (int)tsc[0]);
    __syncthreads();

    u32 thresh; int greater;
    radix_select(row, dead, hist, bc, G_, AUXK, &thresh, &greater);
    const int needEq = AUXK - greater;

    const int base = t * SEG;
    int eqCnt = 0;
    for (int i = 0; i < SEG; ++i) eqCnt += (load_key(row, dead, base + i) == thresh);
    tsc[t] = (u32)eqCnt; excl_scan(tsc);
    const int eqBase = (int)tsc[t];
    __syncthreads();

    int selCnt = 0;
    { int e = eqBase;
      for (int i = 0; i < SEG; ++i) {
          u32 kv = load_key(row, dead, base + i);
          if (kv > thresh) selCnt++;
          else if (kv == thresh) { if (e < needEq) selCnt++; e++; }
      } }
    tsc[t] = (u32)selCnt; excl_scan(tsc);
    int slot = (int)tsc[t];

    { int e = eqBase;
      for (int i = 0; i < SEG; ++i) {
          u32 kv = load_key(row, dead, base + i);
          bool sel = false;
          if (kv > thresh) sel = true;
          else if (kv == thresh) { sel = (e < needEq); e++; }
          if (sel) {
              const float v = dead[base + i] ? row[base + i] : 0.0f;
              aidx[(size_t)b * AUXK + slot] = (u32)(base + i);
              aval[(size_t)b * AUXK + slot] = fmaxf(v, 0.0f);   // relu(top_dead)
              slot++;
          }
      } }
    __syncthreads();

    // overwrite preact row with final feats (zeros except selected positives)
    const u32* bmrow = bitmap + (size_t)b * (G_ / 32);
    for (int i = t; i < G_; i += 256) {
        const float f = row[i];
        const u32 w = bmrow[i >> 5];
        row[i] = ((w >> (i & 31)) & 1u) ? fmaxf(f, 0.0f) : 0.0f;
    }
}

// sparse decode: out[b,o] = (sum_j val_j * Wdec[idx_j, o] + b_dec[o]) * std[b] + mu[b]
__global__ __launch_bounds__(256)
void k_decode(const u32* __restrict__ idx, const float* __restrict__ val, int K,
              const float* __restrict__ Wdec, const float* __restrict__ bdec,
              const float* __restrict__ mu, const float* __restrict__ sd,
              float* __restrict__ outp) {
    __shared__ u32   sidx[AUXK];
    __shared__ float sval[AUXK];
    const int b = blockIdx.y;
    const int o = blockIdx.x * 256 + threadIdx.x;
    for (int j = threadIdx.x; j < K; j += 256) {
        sidx[j] = idx[(size_t)b * K + j];
        sval[j] = val[(size_t)b * K + j];
    }
    __syncthreads();
    float acc = 0.f;
    for (int j = 0; j < K; ++j) {
        const float v = sval[j];
        if (v != 0.f) acc += v * Wdec[(size_t)sidx[j] * O_ + o];
    }
    outp[(size_t)b * O_ + o] = (acc + bdec[o]) * sd[b] + mu[b];
}

__global__ void k_finalize(const int* __restrict__ counter, float* __restrict__ numdead) {
    numdead[0] = (float)(*counter) / (float)B_;
}

// ---------- launcher ----------
extern "C" void kernel_launch(void* const* d_in, const int* in_sizes, int n_in,
                              void* d_out, int out_size, void* d_ws, size_t ws_size,
                              hipStream_t stream) {
    const float* x     = (const float*)d_in[0];
    const float* Wenc  = (const float*)d_in[1];
    const float* benc  = (const float*)d_in[2];
    const float* Wdec  = (const float*)d_in[3];
    const float* bdec  = (const float*)d_in[4];
    const int*   stats = (const int*)  d_in[5];
    const int*   thr   = (const int*)  d_in[8];   // dead_steps_threshold (topk/auxk fixed)

    float* feats   = (float*)d_out;                         // [B,G] (preact scratch, then feats)
    float* recon   = feats   + (size_t)B_ * G_;             // [B,O]
    float* deadrec = recon   + (size_t)B_ * O_;             // [B,O]
    float* numdead = deadrec + (size_t)B_ * O_;             // [1]

    // workspace partition (all 4-byte units)
    u32* w = (u32*)d_ws;
    float* xn     = (float*)w;              w += (size_t)B_ * D_;
    float* muv    = (float*)w;              w += B_;
    float* sdv    = (float*)w;              w += B_;
    u32*   active = w;                      w += G_;
    u32*   deadm  = w;                      w += G_;
    int*   counter= (int*)w;                w += 8;
    u32*   bitmap = w;                      w += (size_t)B_ * (G_ / 32);
    u32*   fidx   = w;                      w += (size_t)B_ * TOPK;
    float* fval   = (float*)w;              w += (size_t)B_ * TOPK;
    u32*   aidx   = w;                      w += (size_t)B_ * AUXK;
    float* aval   = (float*)w;              w += (size_t)B_ * AUXK;

    k_init<<<(G_ + 255) / 256, 256, 0, stream>>>(active, counter);
    k_layernorm<<<B_, 256, 0, stream>>>(x, xn, muv, sdv);

    dim3 ge(G_ / 128, B_ / 128);
    k_encoder<<<ge, 256, 0, stream>>>(xn, Wenc, benc, feats);

    k_top64<<<B_, 256, 0, stream>>>(feats, active, bitmap, fidx, fval);
    k_dead<<<(G_ + 255) / 256, 256, 0, stream>>>(active, stats, thr, deadm);
    k_aux<<<B_, 256, 0, stream>>>(feats, deadm, bitmap, aidx, aval, counter);

    dim3 gd(O_ / 256, B_);
    k_decode<<<gd, 256, 0, stream>>>(fidx, fval, TOPK, Wdec, bdec, muv, sdv, recon);
    k_decode<<<gd, 256, 0, stream>>>(aidx, aval, AUXK, Wdec, bdec, muv, sdv, deadrec);
    k_finalize<<<1, 1, 0, stream>>>(counter, numdead);
}